// RankingLoss_11012296147722
// MI455X (gfx1250) — compile-verified
//
#include <hip/hip_runtime.h>
#include <cstdint>

// RankingLoss on gfx1250 (MI455X): pure-bandwidth rowwise masked-max reduction.
// 256 MB streamed once -> HBM roofline ~11us @ 23.3 TB/s. No matrix math, so
// the CDNA5 path used here is the async tensor-data movement:
//   global_load_async_to_lds_b128 (ASYNCcnt) double-buffering 4KB rows per wave.

#define C_DIM 1024
#define BLOCK 256
#define WPB   8      // waves per block (wave32)
#define NBLK  1024   // 8192 waves total -> 8 rows per wave at B=65536

#define MARGIN_POS 2.5f
#define MARGIN_NEG 0.5f
#define GAMMA_     2.0f

__device__ __forceinline__ float softplus_f(float x) {
  // log1p(exp(x)), stable form (matches reference within rounding)
  return x > 0.0f ? x + log1pf(expf(-x)) : log1pf(expf(x));
}

// Issue one full 4KB row (1024 fp32) as 8 async B128 transfers:
// each lane moves 16B per instruction -> 512B per wave per instruction.
__device__ __forceinline__ void issue_row_async(const float* __restrict__ rowp,
                                                uint32_t lds_byte_base, int lane) {
  uint64_t g = (uint64_t)(uintptr_t)rowp + (uint32_t)(lane * 16);
  uint32_t l = lds_byte_base + (uint32_t)(lane * 16);
#pragma unroll
  for (int c = 0; c < 8; ++c) {
    asm volatile("global_load_async_to_lds_b128 %0, %1, off th:TH_LOAD_NT"
                 :: "v"(l + (uint32_t)(c * 512)),
                    "v"(g + (uint64_t)(c * 512))
                 : "memory");
  }
}

__global__ __launch_bounds__(BLOCK) void ranking_loss_rows(
    const float* __restrict__ scores, const int* __restrict__ labels,
    float* __restrict__ partials, int B) {
  // Per-wave double buffer: 2 x 256 float4 (4KB) per wave -> 64KB per block.
  __shared__ float4 tile[WPB * 2 * 256];
  __shared__ float  wsum[WPB];

  const int lane   = threadIdx.x & 31;   // wave32
  const int wave   = threadIdx.x >> 5;
  const int gwave  = blockIdx.x * WPB + wave;
  const int stride = gridDim.x * WPB;

  // LDS byte offset of this wave's buffers (low 32 bits of flat addr == LDS offset)
  const uint32_t lds_base0 = (uint32_t)(uintptr_t)&tile[(wave * 2 + 0) * 256];
  const uint32_t lds_base1 = (uint32_t)(uintptr_t)&tile[(wave * 2 + 1) * 256];

  float acc = 0.0f;
  int  row  = gwave;
  bool have = row < B;

  if (have) issue_row_async(scores + (size_t)row * C_DIM, lds_base0, lane);

  int parity = 0;
  while (have) {
    const int  next      = row + stride;
    const bool have_next = next < B;

    if (have_next) {
      issue_row_async(scores + (size_t)next * C_DIM,
                      parity ? lds_base0 : lds_base1, lane);
      // 16 async loads in flight; loads complete in order, so <=8 outstanding
      // guarantees the current row's 8 transfers have landed in LDS.
      asm volatile("s_wait_asynccnt 8" ::: "memory");
    } else {
      asm volatile("s_wait_asynccnt 0" ::: "memory");
    }

    const int label = labels[row];
    const float4* __restrict__ buf = &tile[(wave * 2 + parity) * 256];

    float m  = -INFINITY;   // masked row max
    float sp = -INFINITY;   // positive-class score (exactly one lane captures it)
#pragma unroll
    for (int c = 0; c < 8; ++c) {
      const float4 v  = buf[c * 32 + lane];       // ds_load_b128, conflict-free
      const int    cb = c * 128 + lane * 4;
      { bool p = (cb + 0) == label; sp = p ? v.x : sp; m = (!p && v.x > m) ? v.x : m; }
      { bool p = (cb + 1) == label; sp = p ? v.y : sp; m = (!p && v.y > m) ? v.y : m; }
      { bool p = (cb + 2) == label; sp = p ? v.z : sp; m = (!p && v.z > m) ? v.z : m; }
      { bool p = (cb + 3) == label; sp = p ? v.w : sp; m = (!p && v.w > m) ? v.w : m; }
    }
    // Butterfly reduce across the 32 lanes of the wave.
#pragma unroll
    for (int off = 16; off > 0; off >>= 1) {
      m  = fmaxf(m,  __shfl_xor(m,  off, 32));
      sp = fmaxf(sp, __shfl_xor(sp, off, 32));
    }

    const float sneg = (label == 0) ? 0.0f : m;   // OTHER_ID rows: no negative term
    acc += softplus_f(GAMMA_ * (MARGIN_POS - sp))
         + softplus_f(GAMMA_ * (MARGIN_NEG + sneg));

    row = next; have = have_next; parity ^= 1;
  }

  if (lane == 0) wsum[wave] = acc;
  __syncthreads();
  if (threadIdx.x == 0) {
    float s = 0.0f;
#pragma unroll
    for (int w = 0; w < WPB; ++w) s += wsum[w];
    partials[blockIdx.x] = s;   // deterministic per-block partial
  }
}

__global__ __launch_bounds__(BLOCK) void ranking_loss_finalize(
    const float* __restrict__ partials, float* __restrict__ out,
    int n, float invB) {
  __shared__ float sm[BLOCK];
  float s = 0.0f;
  for (int i = threadIdx.x; i < n; i += BLOCK) s += partials[i];
  sm[threadIdx.x] = s;
  __syncthreads();
  for (int st = BLOCK / 2; st > 0; st >>= 1) {
    if (threadIdx.x < st) sm[threadIdx.x] += sm[threadIdx.x + st];
    __syncthreads();
  }
  if (threadIdx.x == 0) out[0] = sm[0] * invB;
}

extern "C" void kernel_launch(void* const* d_in, const int* in_sizes, int n_in,
                              void* d_out, int out_size, void* d_ws, size_t ws_size,
                              hipStream_t stream) {
  (void)n_in; (void)out_size; (void)ws_size;
  const float* scores = (const float*)d_in[0];
  const int*   labels = (const int*)d_in[1];
  const int    B      = in_sizes[1];          // rows
  float* partials = (float*)d_ws;             // NBLK floats of scratch

  ranking_loss_rows<<<NBLK, BLOCK, 0, stream>>>(scores, labels, partials, B);
  ranking_loss_finalize<<<1, BLOCK, 0, stream>>>(partials, (float*)d_out,
                                                 NBLK, 1.0f / (float)B);
}